// MultilevelROIAligner_10136122818732
// MI455X (gfx1250) — compile-verified
//
#include <hip/hip_runtime.h>
#include <hip/hip_bf16.h>

typedef __attribute__((ext_vector_type(2))) float v2f;
typedef __attribute__((ext_vector_type(8))) float v8f;

#define BATCH 2
#define NBOX  512
#define NCH   256
#define CROP  7
#define NSLOT 14            // 2*CROP corner slots per axis
#define KTOT  196           // NSLOT*NSLOT flattened corner-pair slots
#define NPIX  49            // CROP*CROP
#define KSTEPS 49           // KTOT / 4

__device__ __forceinline__ float ldg_off(const float* __restrict__ base, int byteOff) {
    // uniform base + sext(i32) vgpr offset -> global_load saddr form
    return *(const float*)((const char*)base + (long)byteOff);
}

// One workgroup = one (level, batch, box). 512 threads = 16 waves.
// Wave w owns channel tile [16w, 16w+16). Out[pix, ch] = W[pix, rs] * Fsub[rs, ch]
// computed transposed as D(ch x pix) = A(ch x 4rs) * B(4rs x pix) via
// v_wmma_f32_16x16x4_f32, accumulating 49 K-steps over rs = 0..195.
__global__ __launch_bounds__(512)
void roialign_wmma_kernel(const float* __restrict__ f2, const float* __restrict__ f3,
                          const float* __restrict__ f4, const float* __restrict__ f5,
                          const float* __restrict__ boxes, float* __restrict__ out) {
    // interleaved weights: s_wt[(k*64 + pix)*4 + sub] = W[rs = 4k+sub -> pix]
    // so each WMMA B operand (two consecutive subs) is one aligned ds_load_b64
    __shared__ float s_wt[KSTEPS * 64 * 4];
    __shared__ int   s_off[KTOT + 4];      // row byte offsets; padded for prefetch
    __shared__ float s_ky[NSLOT], s_kx[NSLOT];
    __shared__ int   s_Y[NSLOT], s_X[NSLOT];

    const int gid = blockIdx.x;                     // [lvl][b][n]
    const int lvl = gid / (BATCH * NBOX);
    const int rem = gid % (BATCH * NBOX);
    const int b   = rem / NBOX;
    const int n   = rem % NBOX;
    const int Hf  = 256 >> lvl;                     // square feature map
    const float inv_scale = 1.0f / (float)(4 << lvl);
    const float* feat = (lvl == 0) ? f2 : (lvl == 1) ? f3 : (lvl == 2) ? f4 : f5;

    const int tid = threadIdx.x;

    // ---- per-box setup: 14 corner slots per axis (i = slot>>1, p = slot&1) ----
    if (tid < NSLOT) {
        const float* bx = boxes + ((size_t)(b * NBOX + n)) * 4;
        float y1 = bx[0] * inv_scale, x1 = bx[1] * inv_scale;
        float y2 = bx[2] * inv_scale, x2 = bx[3] * inv_scale;
        int i = tid >> 1, p = tid & 1;
        float t  = ((float)i + 0.5f) / (float)CROP;
        float gy = y1 + t * (y2 - y1);
        float gx = x1 + t * (x2 - x1);
        float y0 = fmaxf(floorf(gy), 0.0f);
        float x0 = fmaxf(floorf(gx), 0.0f);
        float ly = gy - y0, lx = gx - x0;
        s_ky[tid] = p ? ly : (1.0f - ly);
        s_kx[tid] = p ? lx : (1.0f - lx);
        float cmax = (float)(Hf - 1);
        s_Y[tid] = (int)fminf(y0 + (float)p, cmax);
        s_X[tid] = (int)fminf(x0 + (float)p, cmax);
    }
    __syncthreads();

    // gather byte-offsets: rs = r*14 + s -> (Y[r]*Hf + X[s]) * NCH * 4  (< 2^27)
    for (int rs = tid; rs < KTOT + 4; rs += 512)
        s_off[rs] = (rs < KTOT) ? (s_Y[rs / NSLOT] * Hf + s_X[rs % NSLOT]) * (NCH * 4) : 0;

    // interleaved weight matrix; pad pixels 49..63 zeroed (no tile-3 predication)
    for (int idx = tid; idx < KSTEPS * 64 * 4; idx += 512) {
        int sub = idx & 3;
        int pix = (idx >> 2) & 63;
        int k   = idx >> 8;
        int rs  = 4 * k + sub;
        int r = rs / NSLOT, s = rs % NSLOT;
        int i = pix / CROP, j = pix % CROP;
        float w = (pix < NPIX && (r >> 1) == i && (s >> 1) == j) ? s_ky[r] * s_kx[s] : 0.0f;
        s_wt[idx] = w;
    }
    __syncthreads();

    // ---- WMMA main loop ----
    const int wv     = tid >> 5;        // channel tile 0..15
    const int lane   = tid & 31;
    const int lane16 = lane & 15;
    const int khi    = lane >> 4;       // lane-half selects K pair {0,1} vs {2,3}
    const int kk     = khi * 2;
    const int chByte = (wv * 16 + lane16) * 4;   // A-matrix M = lane&15

    const float* fbase = feat + (size_t)b * Hf * Hf * NCH;   // uniform base
    const float* wbase = s_wt + lane16 * 4 + kk;             // per-lane B base

    v8f a0 = {}, a1 = {}, a2 = {}, a3 = {};      // 4 pixel tiles

    // software-pipelined A gather (s_off padded so the k=48 prefetch is in-bounds)
    float nA0 = ldg_off(fbase, s_off[kk] + chByte);
    float nA1 = ldg_off(fbase, s_off[kk + 1] + chByte);

    for (int k = 0; k < KSTEPS; ++k) {
        v2f A;
        A.x = nA0;
        A.y = nA1;
        const int nrs = 4 * (k + 1) + kk;                    // prefetch next step
        nA0 = ldg_off(fbase, s_off[nrs] + chByte);
        nA1 = ldg_off(fbase, s_off[nrs + 1] + chByte);

        const float* wr = wbase + k * 256;                   // this k's weight block
        v2f B0 = *(const v2f*)(wr + 0);                      // pixels  0..15
        v2f B1 = *(const v2f*)(wr + 64);                     // pixels 16..31
        v2f B2 = *(const v2f*)(wr + 128);                    // pixels 32..47
        v2f B3 = *(const v2f*)(wr + 192);                    // pixel 48 + zero pad
        a0 = __builtin_amdgcn_wmma_f32_16x16x4_f32(false, A, false, B0, (short)0, a0, false, false);
        a1 = __builtin_amdgcn_wmma_f32_16x16x4_f32(false, A, false, B1, (short)0, a1, false, false);
        a2 = __builtin_amdgcn_wmma_f32_16x16x4_f32(false, A, false, B2, (short)0, a2, false, false);
        a3 = __builtin_amdgcn_wmma_f32_16x16x4_f32(false, A, false, B3, (short)0, a3, false, false);
    }

    // ---- store: D layout gives each lane pixel=lane16, 8 contiguous channels ----
    float* ob = out + ((long)gid * NPIX) * NCH + wv * 16 + khi * 8;
    {
        float* p = ob + (long)(0 * 16 + lane16) * NCH;
        float4 lo; lo.x = a0[0]; lo.y = a0[1]; lo.z = a0[2]; lo.w = a0[3];
        float4 hi; hi.x = a0[4]; hi.y = a0[5]; hi.z = a0[6]; hi.w = a0[7];
        *(float4*)p = lo; *(float4*)(p + 4) = hi;
    }
    {
        float* p = ob + (long)(1 * 16 + lane16) * NCH;
        float4 lo; lo.x = a1[0]; lo.y = a1[1]; lo.z = a1[2]; lo.w = a1[3];
        float4 hi; hi.x = a1[4]; hi.y = a1[5]; hi.z = a1[6]; hi.w = a1[7];
        *(float4*)p = lo; *(float4*)(p + 4) = hi;
    }
    {
        float* p = ob + (long)(2 * 16 + lane16) * NCH;
        float4 lo; lo.x = a2[0]; lo.y = a2[1]; lo.z = a2[2]; lo.w = a2[3];
        float4 hi; hi.x = a2[4]; hi.y = a2[5]; hi.z = a2[6]; hi.w = a2[7];
        *(float4*)p = lo; *(float4*)(p + 4) = hi;
    }
    if (lane16 == 0) {  // pixel 48 only
        float* p = ob + (long)48 * NCH;
        float4 lo; lo.x = a3[0]; lo.y = a3[1]; lo.z = a3[2]; lo.w = a3[3];
        float4 hi; hi.x = a3[4]; hi.y = a3[5]; hi.z = a3[6]; hi.w = a3[7];
        *(float4*)p = lo; *(float4*)(p + 4) = hi;
    }
}

extern "C" void kernel_launch(void* const* d_in, const int* in_sizes, int n_in,
                              void* d_out, int out_size, void* d_ws, size_t ws_size,
                              hipStream_t stream) {
    (void)in_sizes; (void)n_in; (void)d_ws; (void)ws_size; (void)out_size;
    const float* f2    = (const float*)d_in[0];
    const float* f3    = (const float*)d_in[1];
    const float* f4    = (const float*)d_in[2];
    const float* f5    = (const float*)d_in[3];
    const float* boxes = (const float*)d_in[4];
    float* out = (float*)d_out;

    dim3 grid(4 * BATCH * NBOX);   // 4096 workgroups: one per (level, b, box)
    dim3 block(512);               // 16 waves
    roialign_wmma_kernel<<<grid, block, 0, stream>>>(f2, f3, f4, f5, boxes, out);
}